// IPTConv_24051816857901
// MI455X (gfx1250) — compile-verified
//
#include <hip/hip_runtime.h>

// GEMM view of pixel-unshuffle(2) + 1x1 conv:
//   out[o, p] = sum_k W[o,k] * xs[k,p] + bias[o]
//   x:[16,64,256,256] f32, W:[128,256], bias:[128], out:[16,128,128,128]
// Each wave computes full-M (128) x 32 positions (2 N-tiles) with
// V_WMMA_F32_16X16X4_F32: x streamed from HBM exactly once, each LDS A
// fragment feeds 2 WMMAs, accumulators are initialized with the bias.

typedef __attribute__((ext_vector_type(2))) float v2f;
typedef __attribute__((ext_vector_type(8))) float v8f;

#define C_IN   64
#define K_DIM  256          // C_IN * 4
#define M_DIM  128          // output channels
#define HW_IN  256
#define HW_OUT 128
#define BATCH  16
#define WAVES_PER_WG 4
#define NT     2            // N-tiles (of 16 positions) per wave

__global__ __launch_bounds__(WAVES_PER_WG * 32)
void iptconv_wmma_f32(const float* __restrict__ x,
                      const float* __restrict__ W,
                      const float* __restrict__ bias,
                      float* __restrict__ out)
{
    // ---- stage W (128x256 f32 = 128 KB) into LDS, reused 8192x ----
    __shared__ float sW[M_DIM * K_DIM];

    const int tid = threadIdx.x;
    {
        const float4* W4  = (const float4*)W;
        float4*       sW4 = (float4*)sW;
        #pragma unroll
        for (int i = 0; i < (M_DIM * K_DIM / 4) / (WAVES_PER_WG * 32); ++i)
            sW4[i * (WAVES_PER_WG * 32) + tid] = W4[i * (WAVES_PER_WG * 32) + tid];
    }
    __syncthreads();

    const int lane = tid & 31;
    const int wave = tid >> 5;
    const int half = lane >> 4;   // 0: K0/K1 (row 2h),  1: K2/K3 (row 2h+1)
    const int li   = lane & 15;   // M-row (A) / N-col (B,C,D) within half

    // ---- which 32-wide slab of output positions this wave owns ----
    const int np = blockIdx.x * WAVES_PER_WG + wave;   // 0 .. 8191
    const int w0 = (np & 3) * (16 * NT);               // 128/32 = 4 slabs per row
    const int hb = np >> 2;
    const int h  = hb & (HW_OUT - 1);
    const int b  = hb >> 7;

    // B-fragment base: x[b][c][2h+half][2*(w0+li) .. +1]  (contiguous float2)
    const float* xptr = x
        + ((size_t)b * C_IN) * (HW_IN * HW_IN)
        + (size_t)(2 * h + half) * HW_IN
        + (size_t)(2 * (w0 + li));

    // ---- accumulators initialized with bias fragments ----
    // D/C layout: VGPR r, lanes<16 -> M=r, lanes>=16 -> M=r+8
    v8f acc[NT][8];
    #pragma unroll
    for (int mt = 0; mt < 8; ++mt) {
        const float4* bp = (const float4*)(bias + mt * 16 + 8 * half);
        const float4 b0 = bp[0];
        const float4 b1 = bp[1];
        const v8f bf = (v8f){b0.x, b0.y, b0.z, b0.w, b1.x, b1.y, b1.z, b1.w};
        #pragma unroll
        for (int t = 0; t < NT; ++t) acc[t][mt] = bf;
    }

    // A-fragment LDS offset for k-step kk: W[mt*16+li][kk + 2*half .. +1]
    const int aoff = li * K_DIM + 2 * half;

    #pragma unroll 2
    for (int c = 0; c < C_IN; ++c) {
        const float* xc = xptr + (size_t)c * (HW_IN * HW_IN);
        const v2f bfrag0 = *(const v2f*)(xc);
        const v2f bfrag1 = *(const v2f*)(xc + 32);     // +16 positions (128 B)
        const int kk = 4 * c;
        #pragma unroll
        for (int mt = 0; mt < 8; ++mt) {
            const v2f afrag = *(const v2f*)&sW[mt * 16 * K_DIM + aoff + kk];
            acc[0][mt] = __builtin_amdgcn_wmma_f32_16x16x4_f32(
                false, afrag, false, bfrag0, (short)0, acc[0][mt], false, false);
            acc[1][mt] = __builtin_amdgcn_wmma_f32_16x16x4_f32(
                false, afrag, false, bfrag1, (short)0, acc[1][mt], false, false);
        }
    }

    // ---- epilogue: D VGPR r, lane-half -> out[b][mt*16 + r + 8*half][h][w+li]
    #pragma unroll
    for (int t = 0; t < NT; ++t) {
        const int wbase = w0 + t * 16 + li;
        #pragma unroll
        for (int mt = 0; mt < 8; ++mt) {
            #pragma unroll
            for (int r = 0; r < 8; ++r) {
                const int o = mt * 16 + r + 8 * half;
                out[(((size_t)b * M_DIM + o) * HW_OUT + h) * HW_OUT + wbase] =
                    acc[t][mt][r];
            }
        }
    }
}

extern "C" void kernel_launch(void* const* d_in, const int* in_sizes, int n_in,
                              void* d_out, int out_size, void* d_ws, size_t ws_size,
                              hipStream_t stream)
{
    const float* x    = (const float*)d_in[0];   // 16*64*256*256
    const float* W    = (const float*)d_in[1];   // 128*256
    const float* bias = (const float*)d_in[2];   // 128
    float* out        = (float*)d_out;           // 16*128*128*128

    const int n_slabs = (BATCH * HW_OUT * HW_OUT) / (16 * NT);  // 8192
    const int blocks  = n_slabs / WAVES_PER_WG;                 // 2048
    iptconv_wmma_f32<<<blocks, WAVES_PER_WG * 32, 0, stream>>>(x, W, bias, out);
}